// scale_and_CDF_69123203661836
// MI455X (gfx1250) — compile-verified
//
#include <hip/hip_runtime.h>

#define DCOLS 256
#define BOUNDF 30.0f
#define BETAF 1e-8f
#define LOGBETA (-18.420680743952367f)

// workspace layout (floats)
#define WS_MESH  0                    // 64 (padded with +inf past index 32)
#define WS_ELMT  64                   // 32
#define WS_IELMT 96                   // 32
#define WS_PDF   128                  // 33*256
#define WS_FREF  (128 + 33 * DCOLS)   // 32*256

typedef __attribute__((ext_vector_type(2))) float v2f;
typedef __attribute__((ext_vector_type(8))) float v8f;

// ---------------------------------------------------------------------------
// Kernel 1: build mesh / elmt / pdf / F_ref tables in workspace.
// ---------------------------------------------------------------------------
__global__ __launch_bounds__(256) void prep_kernel(const float* __restrict__ p,
                                                   float* __restrict__ ws) {
  __shared__ float sMesh[64];
  __shared__ float sElmt[33];
  const int t = threadIdx.x;

  if (t < 64) {
    float mv;
    if (t == 0) {
      mv = 0.0f;
    } else if (t == 32) {
      mv = 1.0f;
    } else if (t < 32) {
      // double-precision geometric mesh, matching the NumPy float64 build
      const int idx = t - 16;
      const int a = idx < 0 ? -idx : idx;
      double rp = 1.0;
      for (int i = 0; i < a; ++i) rp *= 1.2;
      double r16 = 1.0;
      for (int i = 0; i < 16; ++i) r16 *= 1.2;
      const double x1L = 30.0 * (1.2 - 1.0) / (r16 - 1.0);
      const double tt = (1.0 - rp) / (1.0 - 1.2);
      const double xr = (idx >= 0) ? (x1L * tt) : -(x1L * tt);
      mv = (float)((xr + 30.0) / 60.0);
    } else {
      mv = __builtin_inff();  // pad for branchless binary search
    }
    sMesh[t] = mv;
    ws[WS_MESH + t] = mv;
  }
  __syncthreads();

  if (t < 32) {
    const float e = sMesh[t + 1] - sMesh[t];
    sElmt[t] = e;
    ws[WS_ELMT + t] = e;
    ws[WS_IELMT + t] = 1.0f / e;
  }
  __syncthreads();

  // one thread per column d
  const int d = t;
  float ep[31];
  float s = 0.0f;
#pragma unroll
  for (int i = 0; i < 31; ++i) {
    ep[i] = expf(p[i * DCOLS + d]);
    s += ep[i] * (0.5f * (sElmt[i] + sElmt[i + 1]));
  }
  const float norm = (1.0f - (sElmt[0] + sElmt[31]) * (BETAF * 0.5f)) / s;

  float prev = BETAF;
  float F = 0.0f;
  ws[WS_PDF + 0 * DCOLS + d] = BETAF;
  ws[WS_FREF + 0 * DCOLS + d] = 0.0f;
#pragma unroll
  for (int i = 0; i < 32; ++i) {
    const float cur = (i < 31) ? (norm * ep[i]) : BETAF;
    ws[WS_PDF + (i + 1) * DCOLS + d] = cur;
    const float cell = 0.5f * (prev + cur) * sElmt[i];
    if (i < 31) {
      F += cell;
      ws[WS_FREF + (i + 1) * DCOLS + d] = F;
    }
    prev = cur;
  }
}

// ---------------------------------------------------------------------------
// Kernel 2: elementwise flow transform + WMMA-based per-row logdet reduction.
// Block = 256 threads = 8 waves. Each wave owns 16 rows (one WMMA A tile).
// ---------------------------------------------------------------------------
__global__ __launch_bounds__(256) void flow_kernel(
    const float* __restrict__ x, const float* __restrict__ logdet_in,
    const float* __restrict__ b, const float* __restrict__ logs,
    const float* __restrict__ ws, float* __restrict__ z_out,
    float* __restrict__ ld_out, int batch) {
  __shared__ float sMesh[64];
  __shared__ float sElmt[32];
  __shared__ float sIElmt[32];
  __shared__ float sPdf[33 * DCOLS];

  const int tid = threadIdx.x;
  if (tid < 64) sMesh[tid] = ws[WS_MESH + tid];
  else if (tid < 96) sElmt[tid - 64] = ws[WS_ELMT + (tid - 64)];
  else if (tid < 128) sIElmt[tid - 96] = ws[WS_IELMT + (tid - 96)];
  for (int i = tid; i < 33 * DCOLS; i += 256) sPdf[i] = ws[WS_PDF + i];
  __syncthreads();

  const float* __restrict__ gFref = ws + WS_FREF;  // 32KB table, L2-resident

  const int lane = tid & 31;
  const int wave = tid >> 5;
  const int g = lane >> 4;                     // A-matrix lane group
  const int rowBase = blockIdx.x * 128 + wave * 16;
  const int c0 = lane * 4;                     // fixed columns per lane
  const int c1 = 128 + lane * 4;

  const float4 b0 = *(const float4*)(b + c0);
  const float4 b1 = *(const float4*)(b + c1);
  const float4 l0v = *(const float4*)(logs + c0);
  const float4 l1v = *(const float4*)(logs + c1);
  float4 s0, s1;
  s0.x = expf(l0v.x); s0.y = expf(l0v.y); s0.z = expf(l0v.z); s0.w = expf(l0v.w);
  s1.x = expf(l1v.x); s1.y = expf(l1v.y); s1.z = expf(l1v.z); s1.w = expf(l1v.w);

  // sum(logs): each wave's lanes cover all 256 columns exactly once
  float sl = l0v.x + l0v.y + l0v.z + l0v.w + l1v.x + l1v.y + l1v.z + l1v.w;
#pragma unroll
  for (int m = 16; m >= 1; m >>= 1) sl += __shfl_xor(sl, m, 32);

  auto elem = [&](float xv, float bb, float ss, int d, float& pr) -> float {
    const float xa = (xv + bb) * ss;                      // actnorm
    const float zn = (xa + BOUNDF) * (1.0f / (2.0f * BOUNDF));
    // branchless binary search: k = searchsorted_right(MESH, zn) - 1
    int k = -1;
#pragma unroll
    for (int st = 32; st >= 1; st >>= 1) {
      const int tt = k + st;                              // always in [0,63)
      if (sMesh[tt] <= zn) k = tt;
    }
    const bool cover = (k >= 0) && (k <= 31);
    const int kc = k < 0 ? 0 : (k > 31 ? 31 : k);
    const float xm = zn - sMesh[kc];
    const float v1 = sPdf[kc * DCOLS + d];
    const float v2 = sPdf[kc * DCOLS + DCOLS + d];
    const float slope = (v2 - v1) * sIElmt[kc];
    const float Fp = __ldg(gFref + kc * DCOLS + d);
    const float y = cover ? fmaf(0.5f * xm * xm, slope, fmaf(xm, v1, Fp)) : zn;
    const float dld = cover ? fmaf(xm, slope, v1) : 1.0f;
    float z2 = fmaf(y, 2.0f * BOUNDF, -BOUNDF);
    float lt = logf(dld);
    if (z2 > BOUNDF)  { z2 = fmaf(BETAF, z2 - BOUNDF, BOUNDF);  lt += LOGBETA; }
    if (z2 < -BOUNDF) { z2 = fmaf(BETAF, z2 + BOUNDF, -BOUNDF); lt += LOGBETA; }
    pr += lt;
    return z2;
  };

  // A-matrix accumulators: lane (m + 16g), VGPR j  <->  A[m][2g + j]
  v2f acc;
  acc.x = 0.0f;
  acc.y = 0.0f;

  for (int r = 0; r < 16; ++r) {
    const int row = rowBase + r;
    float pr = 0.0f;
    if (row < batch) {
      const float* xr = x + (size_t)row * DCOLS;
      __builtin_prefetch(xr + DCOLS + c0, 0, 1);  // next row -> global_prefetch
      const float4 X0 = *(const float4*)(xr + c0);
      const float4 X1 = *(const float4*)(xr + c1);
      float4 Z0, Z1;
      Z0.x = elem(X0.x, b0.x, s0.x, c0 + 0, pr);
      Z0.y = elem(X0.y, b0.y, s0.y, c0 + 1, pr);
      Z0.z = elem(X0.z, b0.z, s0.z, c0 + 2, pr);
      Z0.w = elem(X0.w, b0.w, s0.w, c0 + 3, pr);
      Z1.x = elem(X1.x, b1.x, s1.x, c1 + 0, pr);
      Z1.y = elem(X1.y, b1.y, s1.y, c1 + 1, pr);
      Z1.z = elem(X1.z, b1.z, s1.z, c1 + 2, pr);
      Z1.w = elem(X1.w, b1.w, s1.w, c1 + 3, pr);
      *(float4*)(z_out + (size_t)row * DCOLS + c0) = Z0;
      *(float4*)(z_out + (size_t)row * DCOLS + c1) = Z1;
    }
    // fold 32 lane-partials -> 4 chunk sums (lanes [8k,8k+8) all hold S_k)
    pr += __shfl_xor(pr, 1, 32);
    pr += __shfl_xor(pr, 2, 32);
    pr += __shfl_xor(pr, 4, 32);
    const float sA = __shfl(pr, 16 * g, 32);      // S_{2g}
    const float sB = __shfl(pr, 16 * g + 8, 32);  // S_{2g+1}
    if ((lane & 15) == r) { acc.x += sA; acc.y += sB; }
  }

  // D[m][n] = sum_k A[m][k] * 1  -> 16 row sums in one WMMA, full f32
  v2f ones;
  ones.x = 1.0f;
  ones.y = 1.0f;
  v8f cz = {};
  v8f dmat = __builtin_amdgcn_wmma_f32_16x16x4_f32(
      false, acc, false, ones, (short)0, cz, false, false);

  // D layout: VGPR j holds M=j (lanes 0-15) / M=8+j (lanes 16-31)
  const int j = lane & 7;
  float v = dmat[0];
  if (j == 1) v = dmat[1];
  if (j == 2) v = dmat[2];
  if (j == 3) v = dmat[3];
  if (j == 4) v = dmat[4];
  if (j == 5) v = dmat[5];
  if (j == 6) v = dmat[6];
  if (j == 7) v = dmat[7];
  if ((lane & 15) < 8) {
    const int row = rowBase + j + 8 * g;
    if (row < batch) ld_out[row] = logdet_in[row] + sl + v;
  }
}

// ---------------------------------------------------------------------------
extern "C" void kernel_launch(void* const* d_in, const int* in_sizes, int n_in,
                              void* d_out, int out_size, void* d_ws,
                              size_t ws_size, hipStream_t stream) {
  const float* x      = (const float*)d_in[0];  // [BATCH, 256]
  const float* logdet = (const float*)d_in[1];  // [BATCH, 1]
  const float* b      = (const float*)d_in[2];  // [1, 256]
  const float* logs   = (const float*)d_in[3];  // [1, 256]
  const float* p      = (const float*)d_in[4];  // [31, 256]
  float* ws  = (float*)d_ws;
  float* out = (float*)d_out;                   // z then logdet, concatenated
  const int batch = in_sizes[0] / DCOLS;
  float* ldout = out + (size_t)batch * DCOLS;

  prep_kernel<<<1, 256, 0, stream>>>(p, ws);
  const int blocks = (batch + 127) / 128;
  flow_kernel<<<blocks, 256, 0, stream>>>(x, logdet, b, logs, ws, out, ldout,
                                          batch);
}